// DependencyParser_7816840478887
// MI455X (gfx1250) — compile-verified
//
#include <hip/hip_runtime.h>
#include <hip/hip_bf16.h>

typedef float v2f __attribute__((ext_vector_type(2)));
typedef float v8f __attribute__((ext_vector_type(8)));

#define TLEN 769
#define NTOK 768
#define EDIM 256
#define HDIM 128
#define GDIM 512   // 4*H
#define FDIM 128

__device__ __forceinline__ float fexp2(float x) { return __builtin_amdgcn_exp2f(x); }
__device__ __forceinline__ float frcp(float x)  { return __builtin_amdgcn_rcpf(x); }
// sigmoid(x) = 1/(1+e^-x)
__device__ __forceinline__ float fsig(float x) {
    return frcp(1.0f + fexp2(x * -1.442695040888963f));
}
// tanh(x) = 1 - 2/(e^{2x}+1)
__device__ __forceinline__ float ftanh(float x) {
    return 1.0f - 2.0f * frcp(fexp2(x * 2.885390081777927f) + 1.0f);
}

__device__ __forceinline__ v8f wmma4(v2f a, v2f b, v8f c) {
    return __builtin_amdgcn_wmma_f32_16x16x4_f32(false, a, false, b,
                                                 (short)0, c, false, false);
}

// ---------------------------------------------------------------------------
// Build x = concat(sentence_embedded (768x256), root_vec (1x256))
// ---------------------------------------------------------------------------
__global__ void concat_x(const float* __restrict__ se,
                         const float* __restrict__ root,
                         float* __restrict__ x) {
    int idx = blockIdx.x * 256 + threadIdx.x;
    if (idx < TLEN * EDIM) {
        x[idx] = (idx < NTOK * EDIM) ? se[idx] : root[idx - NTOK * EDIM];
    }
}

// ---------------------------------------------------------------------------
// fp32 WMMA GEMM: out[m][n] = sum_k X[m][k] * W[n][koff+k]  (+ bias1[n]+bias2[n])
// Block = 128 threads (4 waves). Each wave: 32 rows x 32 cols
// (2 A-frags x 2 B-frags -> 4 accumulators, 4 WMMAs per 4-wide K step,
//  1 b64 load per WMMA). Block tile: 32 x 128.
// grid.x = N/128, grid.y = ceil(M/32), grid.z = batch.
// ---------------------------------------------------------------------------
__global__ void wmma_gemm(const float* __restrict__ X, int ldx,
                          const float* __restrict__ W, int ldw, int koff,
                          long wStrideZ,
                          const float* __restrict__ bias1,
                          const float* __restrict__ bias2, int biasStrideZ,
                          float* __restrict__ out, int ldo, long outStrideZ,
                          int M, int K) {
    const int lane = threadIdx.x & 31;
    const int wave = threadIdx.x >> 5;
    const int z    = blockIdx.z;
    W   += (size_t)z * wStrideZ;
    out += (size_t)z * outStrideZ;

    const int m0 = blockIdx.y * 32;
    const int n0 = blockIdx.x * 128 + wave * 32;
    const int hi = lane >> 4;     // selects K pair within the fragment
    const int lo = lane & 15;

    int mr0 = m0 + lo;      if (mr0 >= M) mr0 = M - 1;  // clamp loads:
    int mr1 = m0 + 16 + lo; if (mr1 >= M) mr1 = M - 1;  // EXEC stays all-ones
    const float* aptr0 = X + (size_t)mr0 * ldx + 2 * hi;
    const float* aptr1 = X + (size_t)mr1 * ldx + 2 * hi;
    const float* bptr0 = W + (size_t)(n0 + lo)      * ldw + koff + 2 * hi;
    const float* bptr1 = W + (size_t)(n0 + 16 + lo) * ldw + koff + 2 * hi;

    v8f acc00 = {}, acc01 = {}, acc10 = {}, acc11 = {};
#pragma unroll 4
    for (int kb = 0; kb < K; kb += 4) {
        v2f a0 = *(const v2f*)(aptr0 + kb);
        v2f a1 = *(const v2f*)(aptr1 + kb);
        v2f b0 = *(const v2f*)(bptr0 + kb);
        v2f b1 = *(const v2f*)(bptr1 + kb);
        acc00 = wmma4(a0, b0, acc00);
        acc01 = wmma4(a0, b1, acc01);
        acc10 = wmma4(a1, b0, acc10);
        acc11 = wmma4(a1, b1, acc11);
    }

    const int col0 = n0 + lo, col1 = n0 + 16 + lo;
    float bb0 = 0.f, bb1 = 0.f;
    if (bias1) {
        const float* p1 = bias1 + (size_t)z * biasStrideZ;
        bb0 += p1[col0]; bb1 += p1[col1];
    }
    if (bias2) {
        const float* p2 = bias2 + (size_t)z * biasStrideZ;
        bb0 += p2[col0]; bb1 += p2[col1];
    }
#pragma unroll
    for (int r = 0; r < 8; ++r) {
        int row0 = m0 + r + 8 * hi;
        int row1 = m0 + 16 + r + 8 * hi;
        if (row0 < M) {
            out[(size_t)row0 * ldo + col0] = acc00[r] + bb0;
            out[(size_t)row0 * ldo + col1] = acc01[r] + bb1;
        }
        if (row1 < M) {
            out[(size_t)row1 * ldo + col0] = acc10[r] + bb0;
            out[(size_t)row1 * ldo + col1] = acc11[r] + bb1;
        }
    }
}

// ---------------------------------------------------------------------------
// Sequential LSTM scan for one layer, both directions in parallel (grid.x=2).
// pre: [2][T][512] precomputed x@Wih^T+b.  Whh: [2][512][128].
// Lout: [T][256]  (forward -> cols 0..127, backward -> cols 128..255,
// with time reversal folded in).
// Block = 512 threads; h lives in LDS, c in registers of threads 0..127.
// Next timestep's pre row is prefetched (global_prefetch_b8) while the
// current matvec runs -- this scan is the latency-critical path.
// ---------------------------------------------------------------------------
__global__ void lstm_scan(const float* __restrict__ pre,
                          const float* __restrict__ Whh,
                          float* __restrict__ Lout, int T) {
    const int dir = blockIdx.x;
    const int j   = threadIdx.x;                 // 0..511 -> gate row
    const float* preD = pre + (size_t)dir * T * GDIM;
    const float* Wrow = Whh + (size_t)dir * GDIM * HDIM + (size_t)j * HDIM;

    __shared__ __align__(16) float hLds[HDIM];
    __shared__ __align__(16) float gLds[GDIM];
    if (j < HDIM) hLds[j] = 0.f;
    float creg = 0.f;
    __syncthreads();

    for (int s = 0; s < T; ++s) {
        const int t = dir ? (T - 1 - s) : s;
        if (s + 1 < T) {
            const int tn = dir ? (T - 2 - s) : (s + 1);
            __builtin_prefetch(preD + (size_t)tn * GDIM + j, 0, 3);
        }
        // g_j = pre[t][j] + Whh[j,:] . h
        float acc = preD[(size_t)t * GDIM + j];
        const float4* w4 = (const float4*)Wrow;
        const float4* h4 = (const float4*)hLds;
#pragma unroll 8
        for (int k = 0; k < HDIM / 4; ++k) {
            float4 w = w4[k];
            float4 h = h4[k];
            acc = fmaf(w.x, h.x, acc);
            acc = fmaf(w.y, h.y, acc);
            acc = fmaf(w.z, h.z, acc);
            acc = fmaf(w.w, h.w, acc);
        }
        gLds[j] = acc;
        __syncthreads();
        if (j < HDIM) {
            float ig = fsig(gLds[j]);
            float fg = fsig(gLds[j + 128]);
            float gg = ftanh(gLds[j + 256]);
            float og = fsig(gLds[j + 384]);
            creg = fg * creg + ig * gg;
            float hnew = og * ftanh(creg);
            hLds[j] = hnew;
            Lout[(size_t)t * (2 * HDIM) + dir * HDIM + j] = hnew;
        }
        __syncthreads();
    }
}

// ---------------------------------------------------------------------------
// scores[i][j] = sum_f tanh(A[i][f] + B[j][f] + b1[f]) * w2[f] + b2, diag=-inf
// Block = 256 threads -> 16x16 (i,j) tile. A/B tiles staged into LDS with
// async Global->LDS copies (no VGPR transit), waited with s_wait_asynccnt.
// B tile padded to stride 132 to avoid 16-way bank conflicts (64 banks).
// ---------------------------------------------------------------------------
#define BPAD 132
__global__ void scores_kernel(const float* __restrict__ A,
                              const float* __restrict__ Bm,
                              const float* __restrict__ b1,
                              const float* __restrict__ w2,
                              const float* __restrict__ b2,
                              float* __restrict__ out, int NI, int NJ) {
    __shared__ __align__(16) float As[16 * FDIM];
    __shared__ __align__(16) float Bs[16 * BPAD];
    __shared__ float b1s[FDIM];
    __shared__ float w2s[FDIM];

    const int i0 = blockIdx.x * 16;
    const int j0 = blockIdx.y * 16;
    const int tid = threadIdx.x;

    // 512 16B chunks per tile, 2 per thread, copied straight into LDS
    for (int r = tid; r < 16 * FDIM / 4; r += 256) {
        int row = r >> 5;        // /32
        int c4  = r & 31;
        int gi = i0 + row; if (gi >= NI) gi = NI - 1;
        int gj = j0 + row; if (gj >= NJ) gj = NJ - 1;
        unsigned aLds = (unsigned)(size_t)(As + row * FDIM + c4 * 4);
        unsigned bLds = (unsigned)(size_t)(Bs + row * BPAD + c4 * 4);
        const float* aG = A  + (size_t)gi * FDIM + c4 * 4;
        const float* bG = Bm + (size_t)gj * FDIM + c4 * 4;
        asm volatile("global_load_async_to_lds_b128 %0, %1, off"
                     :: "v"(aLds), "v"(aG) : "memory");
        asm volatile("global_load_async_to_lds_b128 %0, %1, off"
                     :: "v"(bLds), "v"(bG) : "memory");
    }
    if (tid < FDIM) { b1s[tid] = b1[tid]; w2s[tid] = w2[tid]; }
    asm volatile("s_wait_asynccnt 0" ::: "memory");
    __syncthreads();

    const int ti = tid >> 4;
    const int tj = tid & 15;
    const float* arow = As + ti * FDIM;
    const float* brow = Bs + tj * BPAD;
    float acc = 0.f;
#pragma unroll 4
    for (int f = 0; f < FDIM; ++f) {
        acc = fmaf(ftanh(arow[f] + brow[f] + b1s[f]), w2s[f], acc);
    }
    const int i = i0 + ti, j = j0 + tj;
    if (i < NI && j < NJ) {
        out[(size_t)i * NJ + j] =
            (i == j) ? -__builtin_huge_valf() : acc + b2[0];
    }
}

// ---------------------------------------------------------------------------
extern "C" void kernel_launch(void* const* d_in, const int* in_sizes, int n_in,
                              void* d_out, int out_size, void* d_ws, size_t ws_size,
                              hipStream_t stream) {
    const float* se   = (const float*)d_in[0];
    // d_in[1] = deps (unused by reference)
    const float* root = (const float*)d_in[2];
    const float* Wih  = (const float*)d_in[3];   // (2,2,512,256)
    const float* Whh  = (const float*)d_in[4];   // (2,2,512,128)
    const float* bih  = (const float*)d_in[5];   // (2,2,512)
    const float* bhh  = (const float*)d_in[6];   // (2,2,512)
    const float* W1   = (const float*)d_in[7];   // (128,512)
    const float* b1   = (const float*)d_in[8];   // (128,)
    const float* w2   = (const float*)d_in[9];   // (128,)
    const float* b2   = (const float*)d_in[10];  // (1,)
    float* out = (float*)d_out;
    float* ws  = (float*)d_ws;

    float* xbuf = ws;                               // 769*256
    float* L0   = xbuf + TLEN * EDIM;               // 769*256
    float* L1   = L0   + TLEN * EDIM;               // 769*256
    float* pre  = L1   + TLEN * EDIM;               // 2*769*512
    float* Abuf = pre  + 2 * TLEN * GDIM;           // 769*128
    float* Bbuf = Abuf + TLEN * FDIM;               // 768*128

    concat_x<<<TLEN, 256, 0, stream>>>(se, root, xbuf);

    for (int l = 0; l < 2; ++l) {
        const float* Xl = (l == 0) ? xbuf : L0;
        float*       Ll = (l == 0) ? L0 : L1;
        // pre[d] = Xl @ Wih[l][d]^T + (bih[l][d] + bhh[l][d]),  d = blockIdx.z
        wmma_gemm<<<dim3(GDIM / 128, (TLEN + 31) / 32, 2), 128, 0, stream>>>(
            Xl, EDIM,
            Wih + (size_t)l * 2 * GDIM * EDIM, EDIM, /*koff=*/0,
            /*wStrideZ=*/(long)GDIM * EDIM,
            bih + (size_t)l * 2 * GDIM, bhh + (size_t)l * 2 * GDIM,
            /*biasStrideZ=*/GDIM,
            pre, GDIM, /*outStrideZ=*/(long)TLEN * GDIM,
            /*M=*/TLEN, /*K=*/EDIM);

        lstm_scan<<<2, GDIM, 0, stream>>>(
            pre, Whh + (size_t)l * 2 * GDIM * HDIM, Ll, TLEN);
    }

    // A = L @ W1[:, :256]^T   (769 x 128)
    wmma_gemm<<<dim3(1, (TLEN + 31) / 32, 1), 128, 0, stream>>>(
        L1, EDIM, W1, /*ldw=*/GDIM, /*koff=*/0, 0,
        nullptr, nullptr, 0, Abuf, FDIM, 0, TLEN, EDIM);
    // B = L[:768] @ W1[:, 256:]^T   (768 x 128)
    wmma_gemm<<<dim3(1, NTOK / 32, 1), 128, 0, stream>>>(
        L1, EDIM, W1, /*ldw=*/GDIM, /*koff=*/EDIM, 0,
        nullptr, nullptr, 0, Bbuf, FDIM, 0, NTOK, EDIM);

    scores_kernel<<<dim3((TLEN + 15) / 16, NTOK / 16), 256, 0, stream>>>(
        Abuf, Bbuf, b1, w2, b2, out, TLEN, NTOK);
}